// EGNNEncoder_12515534701203
// MI455X (gfx1250) — compile-verified
//
#include <hip/hip_runtime.h>
#include <cstddef>

#define NN 50000      // nodes
#define NE 800000     // edges
#define HD 128        // hidden
#define NL 4          // layers
#define LEPS 1e-5f

typedef float v2f __attribute__((ext_vector_type(2)));
typedef float v8f __attribute__((ext_vector_type(8)));

__device__ __forceinline__ float silu_f(float x) {
  return x * (1.0f / (1.0f + __expf(-x)));
}

// ---------------------------------------------------------------------------
// Weight packing: src is [K][HD] row-major. dst is [K/2][HD][2]:
//   dst[p][n][j] = src[2p+j][n]
// so a lane's B fragment {B[k][n], B[k+1][n]} is one aligned b64 load.
// ---------------------------------------------------------------------------
__global__ __launch_bounds__(256) void egnn_pack_kernel(
    const float* __restrict__ src, float* __restrict__ dst, int total) {
  const int tid = blockIdx.x * blockDim.x + threadIdx.x;
  if (tid < total) {
    const int p = tid / (2 * HD);
    const int rem = tid % (2 * HD);
    const int n = rem >> 1, j = rem & 1;
    dst[tid] = src[(2 * p + j) * HD + n];
  }
}

// ---------------------------------------------------------------------------
// Dual-row-tile GEMM chunk: two 16xK A tiles against one shared B, K=128.
// wp: packed weights for this chunk, [64 pairs][HD][2].
// n = lane&15, hs = lane>>4.  Each shared B fragment feeds 2 WMMAs.
// ---------------------------------------------------------------------------
__device__ __forceinline__ void gemm2x_k128(const float* __restrict__ a0,
                                            const float* __restrict__ a1,
                                            const float* __restrict__ wp,
                                            int n, int hs,
                                            v8f accA[8], v8f accB[8]) {
#pragma unroll 2
  for (int kk = 0; kk < 32; ++kk) {
    const int k = kk * 4 + 2 * hs;        // this lane's K-pair start
    const int p = kk * 2 + hs;            // packed pair index
    v2f av0 = *(const v2f*)(a0 + k);
    v2f av1 = *(const v2f*)(a1 + k);
    const float* wrow = wp + (size_t)p * (2 * HD);
#pragma unroll
    for (int ct = 0; ct < 8; ++ct) {
      v2f b = *(const v2f*)(wrow + 2 * (ct * 16 + n));
      accA[ct] = __builtin_amdgcn_wmma_f32_16x16x4_f32(
          false, av0, false, b, (short)0, accA[ct], false, false);
      accB[ct] = __builtin_amdgcn_wmma_f32_16x16x4_f32(
          false, av1, false, b, (short)0, accB[ct], false, false);
    }
  }
}

// Single-row-tile variant (node kernel).
__device__ __forceinline__ void gemm1x_k128(const float* __restrict__ a0,
                                            const float* __restrict__ wp,
                                            int n, int hs, v8f acc[8]) {
#pragma unroll 4
  for (int kk = 0; kk < 32; ++kk) {
    const int k = kk * 4 + 2 * hs;
    const int p = kk * 2 + hs;
    v2f av = *(const v2f*)(a0 + k);
    const float* wrow = wp + (size_t)p * (2 * HD);
#pragma unroll
    for (int ct = 0; ct < 8; ++ct) {
      v2f b = *(const v2f*)(wrow + 2 * (ct * 16 + n));
      acc[ct] = __builtin_amdgcn_wmma_f32_16x16x4_f32(
          false, av, false, b, (short)0, acc[ct], false, false);
    }
  }
}

// h = emb[z]; agg = 0; d2[e] = |pos[row]-pos[col]|^2
__global__ __launch_bounds__(256) void egnn_init_kernel(
    const int* __restrict__ z, const float* __restrict__ emb,
    const float* __restrict__ pos, const int* __restrict__ eidx,
    float* __restrict__ h, float* __restrict__ agg, float* __restrict__ d2) {
  const int tid = blockIdx.x * blockDim.x + threadIdx.x;
  if (tid < NN * HD) {
    const int node = tid / HD, c = tid % HD;
    h[tid] = emb[z[node] * HD + c];
    agg[tid] = 0.0f;
  }
  if (tid < NE) {
    const int r = eidx[tid];
    const int c = eidx[NE + tid];
    const float dx = pos[r * 3 + 0] - pos[c * 3 + 0];
    const float dy = pos[r * 3 + 1] - pos[c * 3 + 1];
    const float dz = pos[r * 3 + 2] - pos[c * 3 + 2];
    d2[tid] = dx * dx + dy * dy + dz * dz;
  }
}

// ---------------------------------------------------------------------------
// Edge kernel: one wave per 32-edge tile (two 16-row WMMA tiles sharing B).
// 4 waves / 128 threads per block. NE % 32 == 0 -> no tail.
// ---------------------------------------------------------------------------
__global__ __launch_bounds__(128) void egnn_edge_kernel(
    const float* __restrict__ h, const float* __restrict__ d2,
    const int* __restrict__ erow, const int* __restrict__ ecol,
    const float* __restrict__ w1p,   // packed [128 pairs][HD][2] (rows 0..255)
    const float* __restrict__ w1raw, // original [257][HD] (for row 256)
    const float* __restrict__ b1,
    const float* __restrict__ w2p,   // packed [64 pairs][HD][2]
    const float* __restrict__ b2,
    float* __restrict__ agg) {
  __shared__ float mst[4][32 * HD];  // per-wave m staging (32 edges x 128)
  __shared__ float d2s[4][32];
  __shared__ int   rws[4][32];

  const int wave = threadIdx.x >> 5;
  const int lane = threadIdx.x & 31;
  const int n  = lane & 15;
  const int hs = lane >> 4;
  const int tile = blockIdx.x * 4 + wave;

  const size_t eA = (size_t)tile * 32 + n;        // row tile A: edges 0..15
  const size_t eB = eA + 16;                      // row tile B: edges 16..31
  const int rA = erow[eA], cA = ecol[eA];
  const int rB = erow[eB], cB = ecol[eB];
  d2s[wave][lane] = d2[(size_t)tile * 32 + lane];
  rws[wave][lane] = erow[(size_t)tile * 32 + lane];

  const float* rowpA = h + (size_t)rA * HD;
  const float* colpA = h + (size_t)cA * HD;
  const float* rowpB = h + (size_t)rB * HD;
  const float* colpB = h + (size_t)cB * HD;

  v8f accA[8], accB[8];
#pragma unroll
  for (int ct = 0; ct < 8; ++ct) {
    accA[ct] = (v8f){0, 0, 0, 0, 0, 0, 0, 0};
    accB[ct] = (v8f){0, 0, 0, 0, 0, 0, 0, 0};
  }

  // GEMM1: features [0:128]=h[row], [128:256]=h[col]
  gemm2x_k128(rowpA, rowpB, w1p, n, hs, accA, accB);
  gemm2x_k128(colpA, colpB, w1p + (size_t)64 * 2 * HD, n, hs, accA, accB);

  // rank-1 d2 term (w1 row 256) + bias + SiLU, stage m into LDS
#pragma unroll
  for (int ct = 0; ct < 8; ++ct) {
    const int col = ct * 16 + n;
    const float wl = w1raw[(size_t)256 * HD + col];
    const float bb = b1[col];
#pragma unroll
    for (int i = 0; i < 8; ++i) {
      const int MA = i + 8 * hs;            // tile A row
      const int MB = MA + 16;               // tile B row
      float vA = accA[ct][i] + d2s[wave][MA] * wl + bb;
      float vB = accB[ct][i] + d2s[wave][MB] * wl + bb;
      mst[wave][MA * HD + col] = silu_f(vA);
      mst[wave][MB * HD + col] = silu_f(vB);
    }
  }

  v8f acc2A[8], acc2B[8];
#pragma unroll
  for (int ct = 0; ct < 8; ++ct) {
    acc2A[ct] = (v8f){0, 0, 0, 0, 0, 0, 0, 0};
    acc2B[ct] = (v8f){0, 0, 0, 0, 0, 0, 0, 0};
  }

  // GEMM2: m (LDS) @ w2
  gemm2x_k128(&mst[wave][n * HD], &mst[wave][(16 + n) * HD], w2p, n, hs,
              acc2A, acc2B);

  // bias + SiLU + atomic scatter into agg[row]
#pragma unroll
  for (int ct = 0; ct < 8; ++ct) {
    const int col = ct * 16 + n;
    const float bb = b2[col];
#pragma unroll
    for (int i = 0; i < 8; ++i) {
      const int MA = i + 8 * hs;
      const int MB = MA + 16;
      const float vA = silu_f(acc2A[ct][i] + bb);
      const float vB = silu_f(acc2B[ct][i] + bb);
      unsafeAtomicAdd(&agg[(size_t)rws[wave][MA] * HD + col], vA);
      unsafeAtomicAdd(&agg[(size_t)rws[wave][MB] * HD + col], vB);
    }
  }
}

// ---------------------------------------------------------------------------
// Node kernel: one wave per 16-node tile; 8 waves per block. NN % 16 == 0.
// ---------------------------------------------------------------------------
__global__ __launch_bounds__(256) void egnn_node_kernel(
    const float* __restrict__ hin, float* __restrict__ agg,
    const float* __restrict__ w1p,  // packed [128 pairs][HD][2]
    const float* __restrict__ b1,
    const float* __restrict__ w2p,  // packed [64 pairs][HD][2]
    const float* __restrict__ b2,
    const float* __restrict__ lng, const float* __restrict__ lnb,
    float* __restrict__ hout) {
  __shared__ float mst[8][16 * HD];

  const int wave = threadIdx.x >> 5;
  const int lane = threadIdx.x & 31;
  const int n  = lane & 15;
  const int hs = lane >> 4;
  const int tile = blockIdx.x * 8 + wave;
  if (tile >= NN / 16) return;   // wave-uniform

  const size_t node = (size_t)tile * 16 + n;
  const float* hb = hin + node * HD;
  const float* ab = agg + node * HD;

  v8f acc[8];
#pragma unroll
  for (int ct = 0; ct < 8; ++ct) acc[ct] = (v8f){0, 0, 0, 0, 0, 0, 0, 0};

  // GEMM1: node_in[0:128]=h, [128:256]=agg
  gemm1x_k128(hb, w1p, n, hs, acc);
  gemm1x_k128(ab, w1p + (size_t)64 * 2 * HD, n, hs, acc);

#pragma unroll
  for (int ct = 0; ct < 8; ++ct) {
    const int col = ct * 16 + n;
    const float bb = b1[col];
#pragma unroll
    for (int i = 0; i < 8; ++i) {
      const int M = i + 8 * hs;
      mst[wave][M * HD + col] = silu_f(acc[ct][i] + bb);
    }
  }

  v8f acc2[8];
#pragma unroll
  for (int ct = 0; ct < 8; ++ct) acc2[ct] = (v8f){0, 0, 0, 0, 0, 0, 0, 0};

  gemm1x_k128(&mst[wave][n * HD], w2p, n, hs, acc2);

  // residual: x = h_old + (acc2 + b2)
  const size_t base = (size_t)tile * 16;
#pragma unroll
  for (int ct = 0; ct < 8; ++ct) {
    const int col = ct * 16 + n;
    const float bb = b2[col];
#pragma unroll
    for (int i = 0; i < 8; ++i) {
      const int M = i + 8 * hs;
      acc2[ct][i] += bb + hin[(base + M) * HD + col];
    }
  }

  // LayerNorm per row M = i + 8*hs; shfl_xor 1/2/4/8 stays inside the half.
#pragma unroll
  for (int i = 0; i < 8; ++i) {
    float s = 0.0f;
#pragma unroll
    for (int ct = 0; ct < 8; ++ct) s += acc2[ct][i];
    s += __shfl_xor(s, 1, 32);
    s += __shfl_xor(s, 2, 32);
    s += __shfl_xor(s, 4, 32);
    s += __shfl_xor(s, 8, 32);
    const float mu = s * (1.0f / HD);
    float q = 0.0f;
#pragma unroll
    for (int ct = 0; ct < 8; ++ct) {
      const float d = acc2[ct][i] - mu;
      q += d * d;
    }
    q += __shfl_xor(q, 1, 32);
    q += __shfl_xor(q, 2, 32);
    q += __shfl_xor(q, 4, 32);
    q += __shfl_xor(q, 8, 32);
    const float inv = rsqrtf(q * (1.0f / HD) + LEPS);
    const int M = i + 8 * hs;
#pragma unroll
    for (int ct = 0; ct < 8; ++ct) {
      const int col = ct * 16 + n;
      const float o = lng[col] * (acc2[ct][i] - mu) * inv + lnb[col];
      hout[(base + M) * HD + col] = o;
      agg[(base + M) * HD + col] = 0.0f;   // ready for next layer
    }
  }
}

extern "C" void kernel_launch(void* const* d_in, const int* in_sizes, int n_in,
                              void* d_out, int out_size, void* d_ws, size_t ws_size,
                              hipStream_t stream) {
  (void)in_sizes; (void)n_in; (void)out_size; (void)ws_size;
  const int*   z    = (const int*)d_in[0];
  const float* pos  = (const float*)d_in[1];
  const int*   eidx = (const int*)d_in[2];
  const float* emb  = (const float*)d_in[3];
  const float* ew1  = (const float*)d_in[4];   // [L][257][HD]
  const float* eb1  = (const float*)d_in[5];
  const float* ew2  = (const float*)d_in[6];   // [L][HD][HD]
  const float* eb2  = (const float*)d_in[7];
  const float* nw1  = (const float*)d_in[8];   // [L][256][HD]
  const float* nb1  = (const float*)d_in[9];
  const float* nw2  = (const float*)d_in[10];  // [L][HD][HD]
  const float* nb2  = (const float*)d_in[11];
  const float* lng  = (const float*)d_in[12];
  const float* lnb  = (const float*)d_in[13];

  float* h    = (float*)d_ws;                  // [N][HD]
  float* agg  = h + (size_t)NN * HD;           // [N][HD]
  float* d2   = agg + (size_t)NN * HD;         // [E]
  float* e1p  = d2 + NE;                       // [L][128][HD][2]
  float* e2p  = e1p + (size_t)NL * 256 * HD;   // [L][64][HD][2]
  float* n1p  = e2p + (size_t)NL * 128 * HD;   // [L][128][HD][2]
  float* n2p  = n1p + (size_t)NL * 256 * HD;   // [L][64][HD][2]

  const int initBlocks = (NN * HD + 255) / 256;   // also covers NE
  egnn_init_kernel<<<initBlocks, 256, 0, stream>>>(z, emb, pos, eidx, h, agg, d2);

  // Pack weights (once per launch; tiny).
  for (int l = 0; l < NL; ++l) {
    egnn_pack_kernel<<<(256 * HD + 255) / 256, 256, 0, stream>>>(
        ew1 + (size_t)l * 257 * HD, e1p + (size_t)l * 256 * HD, 256 * HD);
    egnn_pack_kernel<<<(128 * HD + 255) / 256, 256, 0, stream>>>(
        ew2 + (size_t)l * HD * HD, e2p + (size_t)l * 128 * HD, 128 * HD);
    egnn_pack_kernel<<<(256 * HD + 255) / 256, 256, 0, stream>>>(
        nw1 + (size_t)l * 256 * HD, n1p + (size_t)l * 256 * HD, 256 * HD);
    egnn_pack_kernel<<<(128 * HD + 255) / 256, 256, 0, stream>>>(
        nw2 + (size_t)l * HD * HD, n2p + (size_t)l * 128 * HD, 128 * HD);
  }

  const int edgeBlocks = (NE / 32) / 4;          // 6250, exact
  const int nodeBlocks = ((NN / 16) + 7) / 8;    // 391

  for (int l = 0; l < NL; ++l) {
    egnn_edge_kernel<<<edgeBlocks, 128, 0, stream>>>(
        h, d2, eidx, eidx + NE,
        e1p + (size_t)l * 256 * HD,
        ew1 + (size_t)l * 257 * HD,
        eb1 + (size_t)l * HD,
        e2p + (size_t)l * 128 * HD,
        eb2 + (size_t)l * HD,
        agg);
    float* hout = (l == NL - 1) ? (float*)d_out : h;
    egnn_node_kernel<<<nodeBlocks, 256, 0, stream>>>(
        h, agg,
        n1p + (size_t)l * 256 * HD, nb1 + (size_t)l * HD,
        n2p + (size_t)l * 128 * HD, nb2 + (size_t)l * HD,
        lng + (size_t)l * HD, lnb + (size_t)l * HD,
        hout);
  }
}